// FFTCore_12171937317490
// MI455X (gfx1250) — compile-verified
//
#include <hip/hip_runtime.h>
#include <math.h>

// ---------------------------------------------------------------------------
// 2^24-point complex FFT (split-plane f32) for MI455X / gfx1250.
//
// Four-step: N = N1*N2 = 4096*4096, n = n1 + 4096*n2, k = k2 + 4096*k1:
//   X[k2 + 4096*k1] = FFT_4096_over_n1( W_N^{n1*k2} * FFT_4096_over_n2(x[n1 + 4096*n2]) )
// Pass 1: column FFTs (over n2) + inter-pass twiddle, store transposed to ws.
// Pass 2: row FFTs (over n1), store to out[k2 + 4096*k1].
//
// Each 4096-pt FFT: 3 in-place radix-16 DIT stages in LDS (base-16 digit
// reversal fused into the async global->LDS fill). The 16-point DFT across a
// tile of 16 butterflies is a 16x16x16 complex GEMM executed as 16x
// V_WMMA_F32_16X16X4_F32 (full-f32 matrix op, matching reference precision).
//
// CDNA5-specific paths used:
//   * v_wmma_f32_16x16x4_f32           (f32 matrix core, wave32)
//   * global_load_async_to_lds_b32     (ASYNCcnt DMA fill, per-lane LDS addr
//                                       carries the digit-reversal scatter)
//   * global_store_async_from_lds_b32  (ASYNCcnt DMA drain, pass 2)
//   * s_wait_asynccnt / s_wait_dscnt   (split dep counters)
//   * 256 KB static LDS per workgroup  (320 KB WGP LDS)
//
// d_ws requirement: 2 * 2^24 * 4 bytes = 128 MB intermediate.
// ---------------------------------------------------------------------------

typedef float v2f __attribute__((ext_vector_type(2)));
typedef float v8f __attribute__((ext_vector_type(8)));

#define FFT_N   (1u << 24)
#define LFFT    4096u        // inner FFT length = 16^3
#define COLS    8u           // sub-FFTs per workgroup (== waves per block)
#define THREADS 256u
#define PI2F    6.283185307179586f

// reverse the three base-16 digits of a 12-bit index
__device__ __forceinline__ unsigned dr16_3(unsigned v) {
    return ((v & 0xFu) << 8) | (v & 0xF0u) | ((v >> 8) & 0xFu);
}

// Workgroup-relative LDS byte offset: low 32 bits of the flat shared-aperture
// address.
__device__ __forceinline__ unsigned lds_off(const void* p) {
    return (unsigned)(uintptr_t)p;
}

// Async DMA: 4 bytes from global to the LDS address in ldsaddr, tracked by ASYNCcnt.
__device__ __forceinline__ void async_g2l_b32(unsigned ldsaddr, const float* g) {
    asm volatile("global_load_async_to_lds_b32 %0, %1, off"
                 :: "v"(ldsaddr), "v"((unsigned long long)(uintptr_t)g)
                 : "memory");
}

// Async DMA: 4 bytes from the LDS address in ldsaddr to global, tracked by ASYNCcnt.
__device__ __forceinline__ void async_l2g_b32(float* g, unsigned ldsaddr) {
    asm volatile("global_store_async_from_lds_b32 %0, %1, off"
                 :: "v"((unsigned long long)(uintptr_t)g), "v"(ldsaddr)
                 : "memory");
}

__device__ __forceinline__ void wait_async0() {
    asm volatile("s_wait_asynccnt 0x0" ::: "memory");
}

__device__ __forceinline__ void wait_ds0() {
    asm volatile("s_wait_dscnt 0x0" ::: "memory");
}

// D(16x16 f32) = A(16x4) * B(4x16) + C   — wave32 matrix op
__device__ __forceinline__ v8f wmma4(v2f a, v2f b, v8f c) {
    // f32 WMMA: NEG[0:1] must be 0 (only C-neg exists), so neg_a/neg_b=false.
    return __builtin_amdgcn_wmma_f32_16x16x4_f32(
        /*neg_a=*/false, a, /*neg_b=*/false, b,
        /*c_mod=*/(short)0, c, /*reuse_a=*/false, /*reuse_b=*/false);
}

// Build the DFT16 matrix W[m][r] = exp(-2*pi*i*m*r/16) in the ISA A-operand
// layout: chunk c covers K = 4c..4c+3; lane<16 holds K = 4c+{0,1}, lane>=16
// holds K = 4c+{2,3}; row M = lane & 15.
__device__ __forceinline__ void build_dft16(unsigned m16, unsigned half,
                                            v2f* ar, v2f* ai) {
#pragma unroll
    for (unsigned c = 0; c < 4; ++c) {
        const float k0 = (float)((c << 2) + (half << 1));
        const float a0 = -PI2F * (float)m16 * k0 * (1.0f / 16.0f);
        const float a1 = -PI2F * (float)m16 * (k0 + 1.0f) * (1.0f / 16.0f);
        float s0, c0f, s1, c1f;
        __sincosf(a0, &s0, &c0f);
        __sincosf(a1, &s1, &c1f);
        v2f tr, ti;
        tr[0] = c0f; tr[1] = c1f;
        ti[0] = s0;  ti[1] = s1;
        ar[c] = tr;  ai[c] = ti;
    }
}

// In-place radix-16 DIT FFT of 4096 complex points in `col` (digit-reversed
// input -> natural output). One wave owns the whole column: same-wave LDS
// ops are in order, so no barriers are needed between stages.
__device__ __forceinline__ void fft4096_wave(float2* __restrict__ col,
                                             unsigned m16, unsigned half,
                                             const v2f* ar, const v2f* ai) {
#pragma unroll 1
    for (unsigned stage = 0; stage < 3; ++stage) {
        const unsigned sh = 4u * stage;         // s = 16^stage
        const unsigned s  = 1u << sh;
        const float tscale = -PI2F / (float)(16u << sh);
#pragma unroll 1
        for (unsigned jt = 0; jt < 16; ++jt) {
            // This lane's GEMM column = butterfly index j (N-dim of the GEMM)
            const unsigned j    = (jt << 4) | m16;
            const unsigned jm   = j & (s - 1u);
            const unsigned base = ((j >> sh) << (sh + 4u)) | jm;
            const float tw = tscale * (float)jm;   // twiddle W_{16s}^{r*jm}

            // Gather + twiddle the B operand (16 x 16 data tile, rows = r).
            // Chunk c: lane<16 holds rows 4c+{0,1}, lane>=16 rows 4c+{2,3}.
            v2f br[4], bi[4], nbi[4];
#pragma unroll
            for (unsigned c = 0; c < 4; ++c) {
                const unsigned r0 = (c << 2) + (half << 1);
                const float2 e0 = col[base + (r0 << sh)];
                const float2 e1 = col[base + ((r0 + 1u) << sh)];
                float s0, c0f, s1, c1f;
                __sincosf(tw * (float)r0, &s0, &c0f);
                __sincosf(tw * (float)(r0 + 1u), &s1, &c1f);
                const float d0r = e0.x * c0f - e0.y * s0;
                const float d0i = e0.x * s0 + e0.y * c0f;
                const float d1r = e1.x * c1f - e1.y * s1;
                const float d1i = e1.x * s1 + e1.y * c1f;
                v2f t;
                t[0] = d0r; t[1] = d1r; br[c] = t;
                t[0] = d0i; t[1] = d1i; bi[c] = t;
                t[0] = -d0i; t[1] = -d1i; nbi[c] = t;
            }

            // Complex GEMM: Cr = Wr*Dr + Wi*(-Di), Ci = Wr*Di + Wi*Dr
            v8f cr = {};
            v8f ci = {};
#pragma unroll
            for (unsigned c = 0; c < 4; ++c) cr = wmma4(ar[c], br[c], cr);
#pragma unroll
            for (unsigned c = 0; c < 4; ++c) cr = wmma4(ai[c], nbi[c], cr);
#pragma unroll
            for (unsigned c = 0; c < 4; ++c) ci = wmma4(ar[c], bi[c], ci);
#pragma unroll
            for (unsigned c = 0; c < 4; ++c) ci = wmma4(ai[c], br[c], ci);

            // Scatter D back in place: VGPR v holds output bin k = v + 8*half
            // of column j (= this lane's own j).
#pragma unroll
            for (unsigned v = 0; v < 8; ++v) {
                const unsigned k = v + (half << 3);
                col[base + (k << sh)] = make_float2(cr[v], ci[v]);
            }
        }
    }
}

__global__ __launch_bounds__(THREADS)
void fft_pass1(const float* __restrict__ xre, const float* __restrict__ xim,
               float* __restrict__ zre, float* __restrict__ zim) {
    __shared__ float2 lds[COLS][LFFT];           // 256 KB (CDNA5: 320KB/WGP)
    const unsigned tid  = threadIdx.x;
    const unsigned lane = tid & 31u;
    const unsigned wave = tid >> 5u;             // wave w owns column w
    const unsigned m16  = lane & 15u;
    const unsigned half = lane >> 4u;
    const unsigned c0   = blockIdx.x * COLS;     // n1 base (contiguous digits)

    // Async DMA fill: 8 adjacent columns, strided rows (lanes 0..7 cover one
    // 32B row segment). The base-16 digit reversal of n2 rides in the per-lane
    // LDS destination address of the async load (no VGPR staging).
    for (unsigned i = tid; i < COLS * LFFT; i += THREADS) {
        const unsigned n2 = i >> 3u, sc = i & 7u;
        const unsigned g  = (n2 << 12u) + c0 + sc;
        const unsigned lo = lds_off(&lds[sc][dr16_3(n2)]);
        async_g2l_b32(lo,      xre + g);   // .x
        async_g2l_b32(lo + 4u, xim + g);   // .y
    }
    wait_async0();          // drain this wave's ASYNCcnt before signaling
    __syncthreads();        // now every wave sees the full LDS image

    v2f ar[4], ai[4];
    build_dft16(m16, half, ar, ai);
    fft4096_wave(&lds[wave][0], m16, half, ar, ai);
    __syncthreads();

    // Inter-pass twiddle W_N^{n1*k2} + transposed store: z[k2*4096 + n1].
    // (Twiddle must run in VALU, so this side keeps the VGPR path.)
    for (unsigned i = tid; i < COLS * LFFT; i += THREADS) {
        const unsigned k2 = i >> 3u, sc = i & 7u;
        const unsigned n1 = c0 + sc;
        const float2 v = lds[sc][k2];
        // n1*k2 < 2^24: exactly representable in f32
        const float ang = -PI2F * (float)(n1 * k2) * (1.0f / 16777216.0f);
        float sn, cs;
        __sincosf(ang, &sn, &cs);
        zre[(k2 << 12u) + n1] = v.x * cs - v.y * sn;
        zim[(k2 << 12u) + n1] = v.x * sn + v.y * cs;
    }
}

__global__ __launch_bounds__(THREADS)
void fft_pass2(const float* __restrict__ zre, const float* __restrict__ zim,
               float* __restrict__ ore, float* __restrict__ oim) {
    __shared__ float2 lds[COLS][LFFT];
    const unsigned tid  = threadIdx.x;
    const unsigned lane = tid & 31u;
    const unsigned wave = tid >> 5u;
    const unsigned m16  = lane & 15u;
    const unsigned half = lane >> 4u;
    const unsigned c0   = blockIdx.x * COLS;     // k2 base

    // Async DMA fill from contiguous ws rows (128B-coalesced source),
    // digit-reversal of n1 fused into the LDS destination address.
    for (unsigned i = tid; i < COLS * LFFT; i += THREADS) {
        const unsigned sc = i >> 12u, n1 = i & 4095u;
        const unsigned g  = ((c0 + sc) << 12u) + n1;
        const unsigned lo = lds_off(&lds[sc][dr16_3(n1)]);
        async_g2l_b32(lo,      zre + g);
        async_g2l_b32(lo + 4u, zim + g);
    }
    wait_async0();
    __syncthreads();

    v2f ar[4], ai[4];
    build_dft16(m16, half, ar, ai);
    fft4096_wave(&lds[wave][0], m16, half, ar, ai);
    wait_ds0();             // async-store engine must see our ds_store scatter
    __syncthreads();

    // Async DMA drain: out[k2 + 4096*k1] straight from LDS (no twiddle here).
    for (unsigned i = tid; i < COLS * LFFT; i += THREADS) {
        const unsigned k1 = i >> 3u, sc = i & 7u;
        const unsigned lo = lds_off(&lds[sc][k1]);
        const unsigned g  = (c0 + sc) + (k1 << 12u);
        async_l2g_b32(ore + g, lo);
        async_l2g_b32(oim + g, lo + 4u);
    }
    wait_async0();          // drain before S_ENDPGM
}

extern "C" void kernel_launch(void* const* d_in, const int* in_sizes, int n_in,
                              void* d_out, int out_size, void* d_ws, size_t ws_size,
                              hipStream_t stream) {
    (void)in_sizes; (void)n_in; (void)out_size; (void)ws_size;
    const float* x   = (const float*)d_in[0];   // shape (2, N): [real | imag]
    const float* xre = x;
    const float* xim = x + FFT_N;
    float* zre = (float*)d_ws;                  // needs 128 MB scratch
    float* zim = zre + FFT_N;
    float* ore = (float*)d_out;
    float* oim = ore + FFT_N;

    const dim3 grid(LFFT / COLS);               // 512 workgroups per pass
    fft_pass1<<<grid, dim3(THREADS), 0, stream>>>(xre, xim, zre, zim);
    fft_pass2<<<grid, dim3(THREADS), 0, stream>>>(zre, zim, ore, oim);
}